// NeighborhoodScorePool_1675037245699
// MI455X (gfx1250) — compile-verified
//
#include <hip/hip_runtime.h>
#include <hip/hip_bf16.h>
#include <math.h>
#include <stdint.h>

typedef __attribute__((ext_vector_type(2))) float v2f;
typedef __attribute__((ext_vector_type(8))) float v8f;
typedef __attribute__((ext_vector_type(4))) int v4i;
typedef unsigned long long u64;
typedef unsigned int u32;
typedef __attribute__((address_space(1))) v4i as1_v4i;   // global int4
typedef __attribute__((address_space(3))) v4i as3_v4i;   // LDS int4

#if defined(__AMDGCN__)
#if __has_builtin(__builtin_amdgcn_global_load_async_to_lds_b128) && __has_builtin(__builtin_amdgcn_s_wait_asynccnt)
#define HAVE_ASYNC_LDS 1
#endif
#endif

#define N_NODES   500003
#define N_EDGES   4000000
#define F_DIM     128
#define C_DIM     64
#define NBH       4
#define NUM_FULL  125000
#define A_FULL    500000
#define NCLUST    125001
#define SENT      125000
#define SORTN     (1u<<19)     // >= N_NODES
#define SORTE     (1u<<22)     // >= N_EDGES
#define SCAN_B    256
#define PADKEY    0xFFFFFFFFFFFFFFFFull
#define SENTPACK  (((u64)SENT<<32)|(u64)SENT)

// d_out layout (floats), reference return order:
#define X_POOL_OFF ((size_t)0)
#define EDGE_OFF   ((size_t)NCLUST*F_DIM)
#define POS_OFF    (EDGE_OFF + (size_t)2*N_EDGES)
#define SCORE_OFF  (POS_OFF + (size_t)NCLUST*3)
#define STD_OFF    (SCORE_OFF + (size_t)N_NODES)

// ---------- init ----------
__global__ void k_init_nodes(float* deg, float* aggt, float* aggpos) {
  int i = blockIdx.x*blockDim.x + threadIdx.x;
  if (i >= N_NODES) return;
  deg[i] = 0.f; aggt[i] = 0.f;
  aggpos[3*i+0] = 0.f; aggpos[3*i+1] = 0.f; aggpos[3*i+2] = 0.f;
}
__global__ void k_fill_edges(float* out) {
  size_t i = (size_t)blockIdx.x*blockDim.x + threadIdx.x;
  if (i >= (size_t)2*N_EDGES) return;
  out[EDGE_OFF + i] = (float)SENT;
}

// ---------- edge pass 1: degree + pos aggregation ----------
__global__ void k_edge_acc(const int* __restrict__ src, const int* __restrict__ dst,
                           const float* __restrict__ pos,
                           float* __restrict__ deg, float* __restrict__ aggpos) {
  int e = blockIdx.x*blockDim.x + threadIdx.x;
  if (e >= N_EDGES) return;
  int s = src[e], d = dst[e];
  atomicAdd(&deg[d], 1.f);
  atomicAdd(&aggpos[3*d+0], pos[3*s+0]);
  atomicAdd(&aggpos[3*d+1], pos[3*s+1]);
  atomicAdd(&aggpos[3*d+2], pos[3*s+2]);
}

// ---------- GNN1 via V_WMMA_F32_16X16X4_F32 ----------
// One wave computes a 16-row tile of h = elu(pos@W1r + (aggpos/deg)@W1n + b1),
// looping over 4 column tiles of 16. K=3 padded to 4 with zeros.
__device__ __forceinline__ float elu1(float xx) {
  return xx > 0.f ? xx : (expf(xx) - 1.f);
}
__global__ void k_gnn1(const float* __restrict__ pos, const float* __restrict__ aggpos,
                       const float* __restrict__ deg,
                       const float* __restrict__ W1r, const float* __restrict__ W1n,
                       const float* __restrict__ b1, float* __restrict__ h) {
  int lane = threadIdx.x & 31;
  int row0 = blockIdx.x * 16;
  int m = lane & 15;
  int row = row0 + m; if (row >= N_NODES) row = N_NODES - 1;
  bool hi = lane >= 16;
  float id = 1.f / fmaxf(deg[row], 1.f);
  // A layout (16x4 f32): lanes0-15 -> K=0(v0),K=1(v1); lanes16-31 -> K=2(v0),K=3(v1)
  v2f a_pos, a_agg;
  if (!hi) {
    a_pos.x = pos[3*row+0];        a_pos.y = pos[3*row+1];
    a_agg.x = aggpos[3*row+0]*id;  a_agg.y = aggpos[3*row+1]*id;
  } else {
    a_pos.x = pos[3*row+2];        a_pos.y = 0.f;
    a_agg.x = aggpos[3*row+2]*id;  a_agg.y = 0.f;
  }
  bool full_tile = (row0 + 16) <= N_NODES;
  for (int ct = 0; ct < 4; ++ct) {
    int col = ct*16 + m;
    // B layout (4x16 f32): N = lane%16; lanes0-15 -> K=0(v0),K=1(v1); lanes16-31 -> K=2(v0),K=3(v1)
    v2f b_r, b_n;
    if (!hi) { b_r.x = W1r[0*C_DIM+col]; b_r.y = W1r[1*C_DIM+col];
               b_n.x = W1n[0*C_DIM+col]; b_n.y = W1n[1*C_DIM+col]; }
    else     { b_r.x = W1r[2*C_DIM+col]; b_r.y = 0.f;
               b_n.x = W1n[2*C_DIM+col]; b_n.y = 0.f; }
    v8f c = {};
    c = __builtin_amdgcn_wmma_f32_16x16x4_f32(false, a_pos, false, b_r, (short)0, c, false, false);
    c = __builtin_amdgcn_wmma_f32_16x16x4_f32(false, a_agg, false, b_n, (short)0, c, false, false);
    float bias = b1[col];
    // C/D layout: VGPR v -> M = v (+8 for lanes 16-31), N = lane%16
    int rbase = row0 + (hi ? 8 : 0);
    if (full_tile) {
      #pragma unroll
      for (int v = 0; v < 8; ++v)
        h[(size_t)(rbase + v)*C_DIM + col] = elu1(c[v] + bias);
    } else {
      #pragma unroll
      for (int v = 0; v < 8; ++v) {
        int rm = rbase + v;
        if (rm < N_NODES) h[(size_t)rm*C_DIM + col] = elu1(c[v] + bias);
      }
    }
  }
}

// ---------- per-node linear terms for GNN2 (linearity: agg(cat)@W2n == agg(cat@W2n)) ----------
__global__ void k_node_t(const float* __restrict__ h, const float* __restrict__ pos,
                         const float* __restrict__ W2r, const float* __restrict__ W2n,
                         const float* __restrict__ b2,
                         float* __restrict__ t, float* __restrict__ basev) {
  int i = blockIdx.x*blockDim.x + threadIdx.x;
  if (i >= N_NODES) return;
  float tr = 0.f, tn = 0.f;
  const float4* hr4 = (const float4*)(h + (size_t)i*C_DIM);
  #pragma unroll
  for (int k = 0; k < C_DIM/4; ++k) {
    float4 v = hr4[k];
    tr += v.x*W2r[4*k+0] + v.y*W2r[4*k+1] + v.z*W2r[4*k+2] + v.w*W2r[4*k+3];
    tn += v.x*W2n[4*k+0] + v.y*W2n[4*k+1] + v.z*W2n[4*k+2] + v.w*W2n[4*k+3];
  }
  for (int k = 0; k < 3; ++k) { float p = pos[3*i+k]; tr += p*W2r[C_DIM+k]; tn += p*W2n[C_DIM+k]; }
  t[i] = tn;
  basev[i] = tr + b2[0];
}
__global__ void k_edge_t(const int* __restrict__ src, const int* __restrict__ dst,
                         const float* __restrict__ t, float* __restrict__ aggt) {
  int e = blockIdx.x*blockDim.x + threadIdx.x;
  if (e >= N_EDGES) return;
  atomicAdd(&aggt[dst[e]], t[src[e]]);
}

// ---------- score + sortable keys ----------
__global__ void k_score_key(const float* __restrict__ basev, const float* __restrict__ aggt,
                            const float* __restrict__ deg, float* __restrict__ out,
                            u64* __restrict__ nkeys) {
  u32 i = blockIdx.x*blockDim.x + threadIdx.x;
  if (i >= SORTN) return;
  if (i < N_NODES) {
    float s = basev[i] + aggt[i] / fmaxf(deg[i], 1.f);
    out[SCORE_OFF + i] = s;
    u32 b = __float_as_uint(s);
    b = (b & 0x80000000u) ? ~b : (b | 0x80000000u);   // total order for floats
    nkeys[i] = ((u64)b << 32) | (u64)i;               // stable argsort
  } else {
    nkeys[i] = PADKEY;
  }
}

// ---------- bitonic sort step (shared by node & edge sorts) ----------
__global__ void k_bitonic(u64* __restrict__ keys, int j, int k) {
  int i = blockIdx.x*blockDim.x + threadIdx.x;
  int ixj = i ^ j;
  if (ixj > i) {
    u64 a = keys[i], b = keys[ixj];
    bool asc = ((i & k) == 0);
    if (asc ? (a > b) : (a < b)) { keys[i] = b; keys[ixj] = a; }
  }
}

// ---------- cluster ids (note: small-nbh cluster ids restart at 0, faithful to source) ----------
__global__ void k_cluster(const u64* __restrict__ nkeys, int* __restrict__ cluster) {
  int i = blockIdx.x*blockDim.x + threadIdx.x;
  if (i >= N_NODES) return;
  int node = (int)(nkeys[i] & 0xFFFFFFFFu);
  int cl = (i < A_FULL) ? (i >> 2) : ((i - A_FULL) / (NBH - 1));
  cluster[node] = cl;
}

// ---------- pooling: x max + pos mean/std(ddof=1); block per cluster ----------
// 4 waves per block; each wave async-copies one gathered 512B row of x into LDS
// (GLOBAL_LOAD_ASYNC_TO_LDS_B128, 16B/lane), then 128 threads reduce max from LDS.
__global__ void k_pool(const u64* __restrict__ nkeys, const float* __restrict__ x,
                       const float* __restrict__ pos, float* __restrict__ out) {
  __shared__ float xs[NBH * F_DIM];
  int g = blockIdx.x, f = threadIdx.x;   // blockDim.x == 128
  int base, cnt;
  if (g < NUM_FULL) { base = g*NBH; cnt = NBH; }
  else              { base = A_FULL + (g - NUM_FULL)*(NBH-1); cnt = NBH-1; }
  int nodes[NBH];
  #pragma unroll
  for (int k = 0; k < NBH; ++k) {
    int kk = k < cnt ? k : cnt - 1;      // duplicate last row when cnt==3 (data unused)
    nodes[k] = (int)(nkeys[base + kk] & 0xFFFFFFFFu);
  }
  float mx = -3.402823466e38f;
#if defined(HAVE_ASYNC_LDS)
  {
    int r = f >> 5;                      // wave id -> row 0..3
    int c = f & 31;                      // 16B chunk within row
    const float* gsrc = x + (size_t)nodes[r]*F_DIM + (size_t)c*4;
    __builtin_amdgcn_global_load_async_to_lds_b128(
        (as1_v4i*)(uintptr_t)gsrc,
        (as3_v4i*)(uintptr_t)&xs[r*F_DIM + c*4], 0, 0);
    __builtin_amdgcn_s_wait_asynccnt(0);
  }
  __syncthreads();
  for (int k = 0; k < cnt; ++k) mx = fmaxf(mx, xs[k*F_DIM + f]);
#else
  for (int k = 0; k < cnt; ++k) mx = fmaxf(mx, x[(size_t)nodes[k]*F_DIM + f]);
#endif
  out[X_POOL_OFF + (size_t)g*F_DIM + f] = mx;
  if (f < 3) {
    float s = 0.f;
    for (int k = 0; k < cnt; ++k) s += pos[3*nodes[k] + f];
    float mean = s / (float)cnt;
    float v = 0.f;
    for (int k = 0; k < cnt; ++k) { float d = pos[3*nodes[k] + f] - mean; v += d*d; }
    out[POS_OFF + (size_t)g*3 + f] = mean;
    out[STD_OFF + (size_t)g*3 + f] = sqrtf(v / (float)(cnt - 1));
  }
}

// ---------- edge pair keys ----------
__global__ void k_edge_keys(const int* __restrict__ src, const int* __restrict__ dst,
                            const int* __restrict__ cluster, u64* __restrict__ ekeys) {
  u32 e = blockIdx.x*blockDim.x + threadIdx.x;
  if (e >= SORTE) return;
  if (e < N_EDGES) {
    if (e + 256 < N_EDGES) { __builtin_prefetch(&src[e+256], 0, 1); __builtin_prefetch(&dst[e+256], 0, 1); }
    u32 cs = (u32)cluster[src[e]], cd = (u32)cluster[dst[e]];
    ekeys[e] = (cs != cd) ? (((u64)cs << 32) | (u64)cd) : SENTPACK;
  } else {
    ekeys[e] = PADKEY;
  }
}

// ---------- unique + compact (flag, two-level scan, scatter) ----------
__device__ __forceinline__ bool uniq_flag(const u64* keys, u32 i) {
  u64 k = keys[i];
  if (k == SENTPACK || k == PADKEY) return false;
  return (i == 0) || (k != keys[i-1]);
}
__global__ void k_scan1(const u64* __restrict__ keys, u32* __restrict__ lexcl, u32* __restrict__ bsums) {
  __shared__ u32 sh[SCAN_B];
  u32 i = blockIdx.x*SCAN_B + threadIdx.x;
  u32 f = uniq_flag(keys, i) ? 1u : 0u;
  sh[threadIdx.x] = f;
  __syncthreads();
  for (int off = 1; off < SCAN_B; off <<= 1) {
    u32 v = (threadIdx.x >= (u32)off) ? sh[threadIdx.x - off] : 0u;
    __syncthreads();
    sh[threadIdx.x] += v;
    __syncthreads();
  }
  lexcl[i] = sh[threadIdx.x] - f;
  if (threadIdx.x == SCAN_B-1) bsums[blockIdx.x] = sh[threadIdx.x];
}
__global__ void k_scan2(u32* bsums, int nb) {
  if (blockIdx.x == 0 && threadIdx.x == 0) {
    u32 acc = 0;
    for (int i = 0; i < nb; ++i) { u32 v = bsums[i]; bsums[i] = acc; acc += v; }
  }
}
__global__ void k_scatter(const u64* __restrict__ keys, const u32* __restrict__ lexcl,
                          const u32* __restrict__ bsums, float* __restrict__ out) {
  u32 i = blockIdx.x*blockDim.x + threadIdx.x;
  if (i >= SORTE) return;
  if (uniq_flag(keys, i)) {
    u32 p = bsums[i / SCAN_B] + lexcl[i];
    if (p < N_EDGES) {
      u64 k = keys[i];
      out[EDGE_OFF + p]           = (float)(u32)(k >> 32);
      out[EDGE_OFF + N_EDGES + p] = (float)(u32)(k & 0xFFFFFFFFu);
    }
  }
}

extern "C" void kernel_launch(void* const* d_in, const int* in_sizes, int n_in,
                              void* d_out, int out_size, void* d_ws, size_t ws_size,
                              hipStream_t stream) {
  const float* x    = (const float*)d_in[0];
  const int*   eidx = (const int*)  d_in[1];
  const float* pos  = (const float*)d_in[2];
  const float* W1r  = (const float*)d_in[3];
  const float* W1n  = (const float*)d_in[4];
  const float* b1   = (const float*)d_in[5];
  const float* W2r  = (const float*)d_in[6];
  const float* W2n  = (const float*)d_in[7];
  const float* b2   = (const float*)d_in[8];
  float* out = (float*)d_out;
  const int* src = eidx;
  const int* dst = eidx + N_EDGES;

  char* ws = (char*)d_ws;
  size_t off = 0;
  auto alloc = [&](size_t bytes) -> char* {
    char* p = ws + off; off = (off + bytes + 15) & ~(size_t)15; return p;
  };
  float* deg    = (float*)alloc((size_t)N_NODES*4);
  float* aggpos = (float*)alloc((size_t)N_NODES*3*4);
  float* h      = (float*)alloc((size_t)N_NODES*C_DIM*4);
  float* t      = (float*)alloc((size_t)N_NODES*4);
  float* basev  = (float*)alloc((size_t)N_NODES*4);
  float* aggt   = (float*)alloc((size_t)N_NODES*4);
  u64*   nkeys  = (u64*)  alloc((size_t)SORTN*8);
  int*   cluster= (int*)  alloc((size_t)N_NODES*4);
  u64*   ekeys  = (u64*)  alloc((size_t)SORTE*8);
  u32*   lexcl  = (u32*)  alloc((size_t)SORTE*4);
  u32*   bsums  = (u32*)  alloc((size_t)(SORTE/SCAN_B)*4);
  (void)ws_size; (void)n_in; (void)in_sizes; (void)out_size;

  const int TB = 256;
  int gN  = (N_NODES + TB - 1) / TB;
  int gE  = (N_EDGES + TB - 1) / TB;
  int g2E = (2*N_EDGES + TB - 1) / TB;

  k_init_nodes<<<gN, TB, 0, stream>>>(deg, aggt, aggpos);
  k_fill_edges<<<g2E, TB, 0, stream>>>(out);

  k_edge_acc<<<gE, TB, 0, stream>>>(src, dst, pos, deg, aggpos);

  k_gnn1<<<(N_NODES + 15) / 16, 32, 0, stream>>>(pos, aggpos, deg, W1r, W1n, b1, h);

  k_node_t<<<gN, TB, 0, stream>>>(h, pos, W2r, W2n, b2, t, basev);
  k_edge_t<<<gE, TB, 0, stream>>>(src, dst, t, aggt);
  k_score_key<<<SORTN/TB, TB, 0, stream>>>(basev, aggt, deg, out, nkeys);

  for (int k = 2; k <= (int)SORTN; k <<= 1)
    for (int j = k >> 1; j > 0; j >>= 1)
      k_bitonic<<<SORTN/TB, TB, 0, stream>>>(nkeys, j, k);

  k_cluster<<<gN, TB, 0, stream>>>(nkeys, cluster);
  k_pool<<<NCLUST, F_DIM, 0, stream>>>(nkeys, x, pos, out);

  k_edge_keys<<<SORTE/TB, TB, 0, stream>>>(src, dst, cluster, ekeys);
  for (int k = 2; k <= (int)SORTE; k <<= 1)
    for (int j = k >> 1; j > 0; j >>= 1)
      k_bitonic<<<SORTE/TB, TB, 0, stream>>>(ekeys, j, k);

  k_scan1<<<SORTE/SCAN_B, SCAN_B, 0, stream>>>(ekeys, lexcl, bsums);
  k_scan2<<<1, 1, 0, stream>>>(bsums, SORTE/SCAN_B);
  k_scatter<<<SORTE/TB, TB, 0, stream>>>(ekeys, lexcl, bsums, out);
}